// SimpleGCN_37022618091719
// MI455X (gfx1250) — compile-verified
//
#include <hip/hip_runtime.h>

typedef __attribute__((ext_vector_type(2))) float v2f;
typedef __attribute__((ext_vector_type(8))) float v8f;

#define IN_C  16
#define HID_C 32
#define OUT_C 16

// ---------- init: dinv <- 1.0 (self-loop degree) ----------
__global__ void init_kernel(float* __restrict__ dinv, int n) {
    int i = blockIdx.x * blockDim.x + threadIdx.x;
    if (i < n) dinv[i] = 1.0f;
}

// ---------- degree accumulation over edges ----------
__global__ void degree_kernel(const long long* __restrict__ dst,
                              float* __restrict__ deg, int e) {
    int i = blockIdx.x * blockDim.x + threadIdx.x;
    if (i < e) atomicAdd(&deg[(int)dst[i]], 1.0f);
}

// ---------- dinv = rsqrt(deg) ----------
__global__ void rsqrt_kernel(float* __restrict__ dinv, int n) {
    int i = blockIdx.x * blockDim.x + threadIdx.x;
    if (i < n) dinv[i] = rsqrtf(dinv[i]);
}

// ---------- GEMM1: h1 = x @ W1 (WMMA f32 16x16x4); also seed agg1 with
// ---------- the self-loop contribution h1 * dinv^2 (replaces zero-init) ----------
__global__ void gemm1_kernel(const float* __restrict__ x, const float* __restrict__ W,
                             const float* __restrict__ dinv,
                             float* __restrict__ h, float* __restrict__ agg, int n) {
    int wave = (blockIdx.x * blockDim.x + threadIdx.x) >> 5;
    int lane = threadIdx.x & 31;
    int ntiles = (n + 15) >> 4;
    if (wave >= ntiles) return;               // wave-uniform: EXEC stays all-ones

    int ln15 = lane & 15;                     // M for A, N for B/C
    int hi   = lane >> 4;                     // 0 or 1
    int koff = hi << 1;                       // K sub-offset: 0 or 2

    int rg = wave * 16 + ln15;
    int rc = rg < n ? rg : n - 1;             // clamp loads, keep EXEC full

    v8f c0 = {};
    v8f c1 = {};
    #pragma unroll
    for (int kb = 0; kb < IN_C / 4; ++kb) {
        int k = kb * 4 + koff;
        v2f a = *(const v2f*)(x + (size_t)rc * IN_C + k);   // A 16x4 layout
        v2f b0, b1;                                          // B 4x16 layout
        b0[0] = W[(k    ) * HID_C + ln15];
        b0[1] = W[(k + 1) * HID_C + ln15];
        b1[0] = W[(k    ) * HID_C + 16 + ln15];
        b1[1] = W[(k + 1) * HID_C + 16 + ln15];
        c0 = __builtin_amdgcn_wmma_f32_16x16x4_f32(false, a, false, b0,
                                                   (short)0, c0, false, false);
        c1 = __builtin_amdgcn_wmma_f32_16x16x4_f32(false, a, false, b1,
                                                   (short)0, c1, false, false);
    }

    int mb = wave * 16 + hi * 8;              // first row this half-wave stores
    if (wave * 16 + 16 <= n) {                // wave-uniform full-tile fast path
        float*       ph = h   + (size_t)mb * HID_C + ln15;
        float*       pa = agg + (size_t)mb * HID_C + ln15;
        const float* pd = dinv + mb;
        #pragma unroll
        for (int v = 0; v < 8; ++v) {
            float di = pd[v];
            float w  = di * di;
            float s0 = c0[v], s1 = c1[v];
            ph[v * HID_C]      = s0;
            ph[v * HID_C + 16] = s1;
            pa[v * HID_C]      = s0 * w;
            pa[v * HID_C + 16] = s1 * w;
        }
    } else {                                  // tail tile: guarded
        #pragma unroll
        for (int v = 0; v < 8; ++v) {
            int m = mb + v;
            if (m < n) {
                float di = dinv[m];
                float w  = di * di;
                h[(size_t)m * HID_C + ln15]        = c0[v];
                h[(size_t)m * HID_C + 16 + ln15]   = c1[v];
                agg[(size_t)m * HID_C + ln15]      = c0[v] * w;
                agg[(size_t)m * HID_C + 16 + ln15] = c1[v] * w;
            }
        }
    }
}

// ---------- edge aggregation, layer 1: agg1[dst] += h1[src] * dinv[s]*dinv[d] ----------
// 4 threads per edge, 8 channels each (2 x float4 gather + 8 float atomics)
__global__ void msg1_kernel(const long long* __restrict__ src,
                            const long long* __restrict__ dst,
                            const float* __restrict__ dinv,
                            const float* __restrict__ h1,
                            float* __restrict__ agg1, int e) {
    int t  = blockIdx.x * blockDim.x + threadIdx.x;
    int ei = t >> 2;
    if (ei >= e) return;
    int part = (t & 3) * 8;
    int s = (int)src[ei];
    int d = (int)dst[ei];
    float w = dinv[s] * dinv[d];
    const float4* hp = (const float4*)(h1 + (size_t)s * HID_C + part);
    float*        op = agg1 + (size_t)d * HID_C + part;
    float4 v0 = hp[0];
    float4 v1 = hp[1];
    atomicAdd(op + 0, v0.x * w); atomicAdd(op + 1, v0.y * w);
    atomicAdd(op + 2, v0.z * w); atomicAdd(op + 3, v0.w * w);
    atomicAdd(op + 4, v1.x * w); atomicAdd(op + 5, v1.y * w);
    atomicAdd(op + 6, v1.z * w); atomicAdd(op + 7, v1.w * w);
}

// ---------- GEMM2: r1 = relu(agg1 + b1) fused into A-load; h2 = r1 @ W2.
// ---------- Stores h2, and seeds out = h2 * dinv^2 + b2 (self-loop + bias),
// ---------- so the later edge atomics accumulate on top of it. ----------
__global__ void gemm2_kernel(const float* __restrict__ agg,
                             const float* __restrict__ b1,
                             const float* __restrict__ W,
                             const float* __restrict__ dinv,
                             const float* __restrict__ b2,
                             float* __restrict__ h2, float* __restrict__ out, int n) {
    int wave = (blockIdx.x * blockDim.x + threadIdx.x) >> 5;
    int lane = threadIdx.x & 31;
    int ntiles = (n + 15) >> 4;
    if (wave >= ntiles) return;

    int ln15 = lane & 15;
    int hi   = lane >> 4;
    int koff = hi << 1;

    int rg = wave * 16 + ln15;
    int rc = rg < n ? rg : n - 1;

    v8f c0 = {};
    #pragma unroll
    for (int kb = 0; kb < HID_C / 4; ++kb) {
        int k = kb * 4 + koff;
        v2f raw = *(const v2f*)(agg + (size_t)rc * HID_C + k);
        v2f a;                                 // fused bias + ReLU (layer-1 epilogue)
        a[0] = fmaxf(raw[0] + b1[k],     0.0f);
        a[1] = fmaxf(raw[1] + b1[k + 1], 0.0f);
        v2f b;
        b[0] = W[(k    ) * OUT_C + ln15];
        b[1] = W[(k + 1) * OUT_C + ln15];
        c0 = __builtin_amdgcn_wmma_f32_16x16x4_f32(false, a, false, b,
                                                   (short)0, c0, false, false);
    }

    int   mb = wave * 16 + hi * 8;
    float bv = b2[ln15];
    if (wave * 16 + 16 <= n) {                 // wave-uniform full-tile fast path
        float*       ph = h2  + (size_t)mb * OUT_C + ln15;
        float*       po = out + (size_t)mb * OUT_C + ln15;
        const float* pd = dinv + mb;
        #pragma unroll
        for (int v = 0; v < 8; ++v) {
            float di = pd[v];
            float s  = c0[v];
            ph[v * OUT_C] = s;
            po[v * OUT_C] = s * di * di + bv;  // self-loop + bias seed
        }
    } else {
        #pragma unroll
        for (int v = 0; v < 8; ++v) {
            int m = mb + v;
            if (m < n) {
                float di = dinv[m];
                float s  = c0[v];
                h2[(size_t)m * OUT_C + ln15]  = s;
                out[(size_t)m * OUT_C + ln15] = s * di * di + bv;
            }
        }
    }
}

// ---------- edge aggregation, layer 2: out[dst] += h2[src] * w ----------
// 4 threads per edge, 4 channels each (1 x float4 gather + 4 float atomics)
__global__ void msg2_kernel(const long long* __restrict__ src,
                            const long long* __restrict__ dst,
                            const float* __restrict__ dinv,
                            const float* __restrict__ h2,
                            float* __restrict__ out, int e) {
    int t  = blockIdx.x * blockDim.x + threadIdx.x;
    int ei = t >> 2;
    if (ei >= e) return;
    int part = (t & 3) * 4;
    int s = (int)src[ei];
    int d = (int)dst[ei];
    float w = dinv[s] * dinv[d];
    float4 v0 = *(const float4*)(h2 + (size_t)s * OUT_C + part);
    float* op = out + (size_t)d * OUT_C + part;
    atomicAdd(op + 0, v0.x * w); atomicAdd(op + 1, v0.y * w);
    atomicAdd(op + 2, v0.z * w); atomicAdd(op + 3, v0.w * w);
}

extern "C" void kernel_launch(void* const* d_in, const int* in_sizes, int n_in,
                              void* d_out, int out_size, void* d_ws, size_t ws_size,
                              hipStream_t stream) {
    const float*     x   = (const float*)d_in[0];
    const float*     W1  = (const float*)d_in[1];
    const float*     b1  = (const float*)d_in[2];
    const float*     W2  = (const float*)d_in[3];
    const float*     b2  = (const float*)d_in[4];
    const long long* ei  = (const long long*)d_in[5];   // int64 edge_index [2, E]

    int n = in_sizes[0] / IN_C;       // 100000
    int e = in_sizes[5] / 2;          // 3200000
    const long long* src = ei;
    const long long* dst = ei + e;

    // workspace layout (floats): dinv[n] | h1[32n] | agg1[32n] | h2[16n]
    float* ws   = (float*)d_ws;
    float* dinv = ws;
    float* h1   = ws + (size_t)n;
    float* agg1 = ws + (size_t)n * 33;
    float* h2   = ws + (size_t)n * 65;
    float* out  = (float*)d_out;

    const int B = 256;
    int ntiles = (n + 15) / 16;
    int gemm_blocks = (ntiles * 32 + B - 1) / B;   // one wave (32 thr) per tile

    init_kernel<<<(n + B - 1) / B, B, 0, stream>>>(dinv, n);
    degree_kernel<<<(e + B - 1) / B, B, 0, stream>>>(dst, dinv, e);
    rsqrt_kernel<<<(n + B - 1) / B, B, 0, stream>>>(dinv, n);

    // layer 1: h1 = x@W1; agg1 seeded with self-loop, then edge scatter-adds
    gemm1_kernel<<<gemm_blocks, B, 0, stream>>>(x, W1, dinv, h1, agg1, n);
    msg1_kernel<<<(e * 4 + B - 1) / B, B, 0, stream>>>(src, dst, dinv, h1, agg1, e);

    // layer 2: fused bias+ReLU -> h2 = r1@W2; out seeded (self-loop + bias),
    // then edge scatter-adds complete the result
    gemm2_kernel<<<gemm_blocks, B, 0, stream>>>(agg1, b1, W2, dinv, b2, h2, out, n);
    msg2_kernel<<<(e * 4 + B - 1) / B, B, 0, stream>>>(src, dst, dinv, h2, out, e);
}